// MiFeatureExtraction_32014686224563
// MI455X (gfx1250) — compile-verified
//
#include <hip/hip_runtime.h>
#include <hip/hip_bf16.h>

// ---------------------------------------------------------------------------
// HSIC pipeline for MI455X (gfx1250), wave32, f32 WMMA 16x16x4.
// M=4096, DIM_X=1024, DIM_Y=128, PROJ=128.
// ---------------------------------------------------------------------------

#define MM        4096
#define DIMX      1024
#define DIMY      128
#define PROJ      128
#define NTILE     (MM / 16)          // 256 tiles per dimension
#define NTILES2   (NTILE * NTILE)    // 65536 output tiles
#define RANK1     8388608u           // (M*M)/2      (1-based)
#define RANK2     8388609u           // (M*M)/2 + 1  (1-based)

typedef float v2f __attribute__((ext_vector_type(2)));
typedef float v8f __attribute__((ext_vector_type(8)));

// ---- WMMA helpers ---------------------------------------------------------
// A 16x4 f32 fragment (ISA 7.12.2): lanes 0-15 hold row M=lane, K={k0,k0+1};
// lanes 16-31 hold row M=lane-16, K={k0+2,k0+3}. B (4x16) mirrors this with
// N=lane%16. For Z (row-major, 128 cols) both A rows and B columns load with
// the identical per-lane pattern, just different base rows.
__device__ __forceinline__ v8f zzt_tile(const float* __restrict__ Z,
                                        int i0, int j0, int lane) {
  const int hl = lane >> 4;          // 0 or 1
  const int lr = lane & 15;
  const float* arow = Z + (size_t)(i0 + lr) * PROJ + 2 * hl;
  const float* brow = Z + (size_t)(j0 + lr) * PROJ + 2 * hl;
  v8f acc = {0.f, 0.f, 0.f, 0.f, 0.f, 0.f, 0.f, 0.f};
#pragma unroll
  for (int k = 0; k < PROJ; k += 4) {
    v2f a = *(const v2f*)(arow + k);
    v2f b = *(const v2f*)(brow + k);
    acc = __builtin_amdgcn_wmma_f32_16x16x4_f32(false, a, false, b,
                                                (short)0, acc, false, false);
  }
  return acc;
}

__device__ __forceinline__ unsigned ordered_u32(float v) {
  unsigned u = __float_as_uint(v);
  return (u & 0x80000000u) ? ~u : (u | 0x80000000u);
}

// ---- utility kernels ------------------------------------------------------
__global__ void zero_kernel(unsigned* p, int n) {
  for (int i = blockIdx.x * blockDim.x + threadIdx.x; i < n;
       i += gridDim.x * blockDim.x)
    p[i] = 0u;
}

// Per-column BN stats -> fused scale/shift: z_n = a[c]*z + b[c]
__global__ void bn_stats_kernel(const float* __restrict__ x,
                                const float* __restrict__ y,
                                const float* gx, const float* bx,
                                const float* gy, const float* by,
                                float* a_x, float* b_x,
                                float* a_y, float* b_y) {
  int c = blockIdx.x * blockDim.x + threadIdx.x;
  if (c < DIMX) {
    double s = 0.0, s2 = 0.0;
    for (int i = 0; i < MM; ++i) {
      float v = x[(size_t)i * DIMX + c];
      s += v; s2 += (double)v * v;
    }
    double mu = s / MM, var = s2 / MM - mu * mu;
    float a = gx[c] * (float)rsqrt(var + 1e-5);
    a_x[c] = a;
    b_x[c] = bx[c] - (float)mu * a;
  } else if (c < DIMX + DIMY) {
    int cc = c - DIMX;
    double s = 0.0, s2 = 0.0;
    for (int i = 0; i < MM; ++i) {
      float v = y[(size_t)i * DIMY + cc];
      s += v; s2 += (double)v * v;
    }
    double mu = s / MM, var = s2 / MM - mu * mu;
    float a = gy[cc] * (float)rsqrt(var + 1e-5);
    a_y[cc] = a;
    b_y[cc] = by[cc] - (float)mu * a;
  }
}

// Fold BN into the projection: W'[p,k] = a_x[k]*W[p,k],
// b'[p] = b[p] + sum_k b_x[k]*W[p,k]
__global__ void prep_w_kernel(const float* __restrict__ W, const float* b,
                              const float* a_x, const float* b_x,
                              float* __restrict__ Wp, float* bp) {
  int p = blockIdx.x, t = threadIdx.x;
  __shared__ float red[256];
  float s = 0.f;
  for (int k = t; k < DIMX; k += 256) {
    float w = W[(size_t)p * DIMX + k];
    Wp[(size_t)p * DIMX + k] = a_x[k] * w;
    s = fmaf(b_x[k], w, s);
  }
  red[t] = s; __syncthreads();
  for (int st = 128; st; st >>= 1) {
    if (t < st) red[t] += red[t + st];
    __syncthreads();
  }
  if (t == 0) bp[p] = b[p] + red[0];
}

// xp = x @ W'^T + b'  (4096 x 128, K=1024) via f32 WMMA
__global__ void proj_kernel(const float* __restrict__ x,
                            const float* __restrict__ Wp,
                            const float* __restrict__ bp,
                            float* __restrict__ xp) {
  int wave = (blockIdx.x * blockDim.x + threadIdx.x) >> 5;
  if (wave >= NTILE * (PROJ / 16)) return;   // wave-uniform guard
  int i0 = (wave >> 3) * 16;
  int j0 = (wave & 7) * 16;
  int lane = threadIdx.x & 31;
  int hl = lane >> 4, lr = lane & 15;
  const float* arow = x  + (size_t)(i0 + lr) * DIMX + 2 * hl;
  const float* brow = Wp + (size_t)(j0 + lr) * DIMX + 2 * hl;
  v8f acc = {0.f, 0.f, 0.f, 0.f, 0.f, 0.f, 0.f, 0.f};
  for (int k = 0; k < DIMX; k += 4) {
    v2f a = *(const v2f*)(arow + k);
    v2f b = *(const v2f*)(brow + k);
    acc = __builtin_amdgcn_wmma_f32_16x16x4_f32(false, a, false, b,
                                                (short)0, acc, false, false);
  }
  int col = j0 + lr;
  float bias = bp[col];
#pragma unroll
  for (int r = 0; r < 8; ++r) {
    int row = i0 + r + hl * 8;
    xp[(size_t)row * PROJ + col] = acc[r] + bias;
  }
}

__global__ void apply_bn_y(const float* __restrict__ y,
                           const float* a_y, const float* b_y,
                           float* __restrict__ yn) {
  int idx = blockIdx.x * blockDim.x + threadIdx.x;
  if (idx < MM * DIMY) {
    int j = idx & (DIMY - 1);
    yn[idx] = fmaf(a_y[j], y[idx], b_y[j]);
  }
}

__global__ void norms_kernel(const float* __restrict__ xp,
                             const float* __restrict__ yn,
                             float* n2x, float* n2y) {
  int i = blockIdx.x * blockDim.x + threadIdx.x;
  if (i < MM) {
    const float* p = xp + (size_t)i * PROJ;
    float s = 0.f;
    for (int k = 0; k < PROJ; ++k) s = fmaf(p[k], p[k], s);
    n2x[i] = s;
  } else if (i < 2 * MM) {
    const float* p = yn + (size_t)(i - MM) * DIMY;
    float s = 0.f;
    for (int k = 0; k < DIMY; ++k) s = fmaf(p[k], p[k], s);
    n2y[i - MM] = s;
  }
}

// ---- median: two-level radix select over ordered float bits ---------------
__global__ void hist_hi_kernel(const float* __restrict__ Z,
                               const float* __restrict__ n2,
                               unsigned* __restrict__ hist) {
  int wave = (blockIdx.x * blockDim.x + threadIdx.x) >> 5;
  if (wave >= NTILES2) return;
  int i0 = (wave >> 8) * 16, j0 = (wave & 255) * 16;
  int lane = threadIdx.x & 31;
  v8f g = zzt_tile(Z, i0, j0, lane);
  int hl = lane >> 4, lr = lane & 15;
  float nj = n2[j0 + lr];
#pragma unroll
  for (int r = 0; r < 8; ++r) {
    float pd = fmaf(-2.f, g[r], n2[i0 + r + hl * 8] + nj);
    atomicAdd(&hist[ordered_u32(pd) >> 16], 1u);
  }
}

__global__ void scan_hi_kernel(const unsigned* __restrict__ hist,
                               unsigned r1, unsigned r2, unsigned* sel) {
  __shared__ unsigned part[256];
  unsigned t = threadIdx.x, s = 0;
  for (int i = 0; i < 256; ++i) s += hist[t * 256 + i];
  part[t] = s; __syncthreads();
  if (t == 0) {
    unsigned ranks[2] = {r1, r2};
    for (int q = 0; q < 2; ++q) {
      unsigned long long cum = 0; unsigned r = ranks[q];
      int seg = 0;
      for (; seg < 255; ++seg) {
        if (cum + part[seg] >= r) break;
        cum += part[seg];
      }
      int bin = seg * 256;
      for (int i = 0; i < 256; ++i) {
        unsigned c = hist[seg * 256 + i];
        if (cum + c >= r) { bin = seg * 256 + i; break; }
        cum += c;
      }
      sel[q * 2 + 0] = (unsigned)bin;
      sel[q * 2 + 1] = (unsigned)(r - cum);
    }
  }
}

__global__ void hist_lo_kernel(const float* __restrict__ Z,
                               const float* __restrict__ n2,
                               const unsigned* __restrict__ sel,
                               unsigned* __restrict__ h2a,
                               unsigned* __restrict__ h2b) {
  int wave = (blockIdx.x * blockDim.x + threadIdx.x) >> 5;
  if (wave >= NTILES2) return;
  int i0 = (wave >> 8) * 16, j0 = (wave & 255) * 16;
  int lane = threadIdx.x & 31;
  v8f g = zzt_tile(Z, i0, j0, lane);
  int hl = lane >> 4, lr = lane & 15;
  unsigned bin1 = sel[0], bin2 = sel[2];
  float nj = n2[j0 + lr];
#pragma unroll
  for (int r = 0; r < 8; ++r) {
    float pd = fmaf(-2.f, g[r], n2[i0 + r + hl * 8] + nj);
    unsigned u = ordered_u32(pd), hi = u >> 16, lo = u & 0xffffu;
    if (hi == bin1) atomicAdd(&h2a[lo], 1u);
    if (hi == bin2) atomicAdd(&h2b[lo], 1u);
  }
}

__global__ void scan_lo_kernel(const unsigned* __restrict__ h2a,
                               const unsigned* __restrict__ h2b,
                               const unsigned* __restrict__ sel,
                               float* sigma) {
  __shared__ unsigned pa[256], pb[256];
  unsigned t = threadIdx.x, sa = 0, sb = 0;
  for (int i = 0; i < 256; ++i) {
    sa += h2a[t * 256 + i];
    sb += h2b[t * 256 + i];
  }
  pa[t] = sa; pb[t] = sb; __syncthreads();
  if (t == 0) {
    float v[2];
    for (int q = 0; q < 2; ++q) {
      const unsigned* h = q ? h2b : h2a;
      const unsigned* pp = q ? pb : pa;
      unsigned bin = sel[q * 2 + 0], r = sel[q * 2 + 1];
      unsigned long long cum = 0;
      int seg = 0;
      for (; seg < 255; ++seg) {
        if (cum + pp[seg] >= r) break;
        cum += pp[seg];
      }
      int lo = seg * 256;
      for (int i = 0; i < 256; ++i) {
        unsigned c = h[seg * 256 + i];
        if (cum + c >= r) { lo = seg * 256 + i; break; }
        cum += c;
      }
      unsigned u = (bin << 16) | (unsigned)lo;
      unsigned f = (u & 0x80000000u) ? (u & 0x7fffffffu) : ~u;
      v[q] = __uint_as_float(f);
    }
    sigma[0] = fmaxf(1e-6f, 0.5f * (v[0] + v[1]));
  }
}

// ---- fused final pass: K, L tiles + all reductions ------------------------
__global__ void hsic_pass_kernel(const float* __restrict__ xp,
                                 const float* __restrict__ yn,
                                 const float* __restrict__ n2x,
                                 const float* __restrict__ n2y,
                                 const float* sigx, const float* sigy,
                                 float* __restrict__ Ksum,
                                 float* __restrict__ Lsum,
                                 double* __restrict__ acc3) {
  int wave = (blockIdx.x * blockDim.x + threadIdx.x) >> 5;
  if (wave >= NTILES2) return;
  int i0 = (wave >> 8) * 16, j0 = (wave & 255) * 16;
  int lane = threadIdx.x & 31;
  v8f gx = zzt_tile(xp, i0, j0, lane);
  v8f gy = zzt_tile(yn, i0, j0, lane);
  int hl = lane >> 4, lr = lane & 15;
  float isx = -1.0f / sigx[0], isy = -1.0f / sigy[0];
  float njx = n2x[j0 + lr], njy = n2y[j0 + lr];
  float colK = 0.f, colL = 0.f, slk = 0.f;
#pragma unroll
  for (int r = 0; r < 8; ++r) {
    int row = i0 + r + hl * 8;
    float pdx = fmaf(-2.f, gx[r], n2x[row] + njx);
    float pdy = fmaf(-2.f, gy[r], n2y[row] + njy);
    float K = __expf(pdx * isx);
    float L = __expf(pdy * isy);
    colK += K; colL += L; slk = fmaf(K, L, slk);
  }
  // K,L symmetric -> row sums == column sums; accumulate column partials.
  atomicAdd(&Ksum[j0 + lr], colK);
  atomicAdd(&Lsum[j0 + lr], colL);
  float sK = colK, sL = colL, sS = slk;
  for (int off = 16; off; off >>= 1) {
    sK += __shfl_xor(sK, off, 32);
    sL += __shfl_xor(sL, off, 32);
    sS += __shfl_xor(sS, off, 32);
  }
  if (lane == 0) {
    atomicAdd(&acc3[0], (double)sS);
    atomicAdd(&acc3[1], (double)sK);
    atomicAdd(&acc3[2], (double)sL);
  }
}

__global__ void finalize_kernel(const float* __restrict__ Ksum,
                                const float* __restrict__ Lsum,
                                const double* __restrict__ acc3,
                                float* out) {
  __shared__ double red[256];
  double d = 0.0;
  for (int i = threadIdx.x; i < MM; i += 256)
    d += (double)Ksum[i] * (double)Lsum[i];
  red[threadIdx.x] = d; __syncthreads();
  for (int s = 128; s; s >>= 1) {
    if (threadIdx.x < s) red[threadIdx.x] += red[threadIdx.x + s];
    __syncthreads();
  }
  if (threadIdx.x == 0) {
    const double m = (double)MM;
    double v = acc3[0] - (2.0 / m) * red[0] + acc3[1] * acc3[2] / (m * m);
    out[0] = (float)(v / ((m - 1.0) * (m - 1.0)));
  }
}

// ---------------------------------------------------------------------------
extern "C" void kernel_launch(void* const* d_in, const int* in_sizes, int n_in,
                              void* d_out, int out_size, void* d_ws,
                              size_t ws_size, hipStream_t stream) {
  const float* x  = (const float*)d_in[0];
  const float* y  = (const float*)d_in[1];
  const float* gx = (const float*)d_in[2];
  const float* bx = (const float*)d_in[3];
  const float* gy = (const float*)d_in[4];
  const float* by = (const float*)d_in[5];
  const float* W  = (const float*)d_in[6];
  const float* b  = (const float*)d_in[7];

  char* ws = (char*)d_ws;
  size_t off = 0;
  auto alloc = [&](size_t bytes) -> char* {
    char* p = ws + off;
    off += (bytes + 255) & ~(size_t)255;
    return p;
  };

  float* a_x = (float*)alloc(DIMX * 4);
  float* b_x = (float*)alloc(DIMX * 4);
  float* a_y = (float*)alloc(DIMY * 4);
  float* b_y = (float*)alloc(DIMY * 4);
  float* bp  = (float*)alloc(PROJ * 4);
  float* Wp  = (float*)alloc((size_t)PROJ * DIMX * 4);
  float* xp  = (float*)alloc((size_t)MM * PROJ * 4);
  float* yn  = (float*)alloc((size_t)MM * DIMY * 4);
  float* n2x = (float*)alloc(MM * 4);
  float* n2y = (float*)alloc(MM * 4);
  // contiguous accumulator region (zeroed in one launch)
  unsigned* hist1 = (unsigned*)alloc(65536 * 4);
  unsigned* h2a   = (unsigned*)alloc(65536 * 4);
  unsigned* h2b   = (unsigned*)alloc(65536 * 4);
  float*    Ksum  = (float*)alloc(MM * 4);
  float*    Lsum  = (float*)alloc(MM * 4);
  double*   acc3  = (double*)alloc(3 * 8);
  unsigned* sel   = (unsigned*)alloc(16 * 4);
  float*    sigx  = (float*)alloc(256);
  float*    sigy  = (float*)alloc(256);

  int zero_words = (int)((((char*)sel) - (char*)hist1) / 4);
  int hist_words = 3 * 65536;

  zero_kernel<<<256, 256, 0, stream>>>(hist1, zero_words);
  bn_stats_kernel<<<(DIMX + DIMY + 255) / 256, 256, 0, stream>>>(
      x, y, gx, bx, gy, by, a_x, b_x, a_y, b_y);
  prep_w_kernel<<<PROJ, 256, 0, stream>>>(W, b, a_x, b_x, Wp, bp);
  proj_kernel<<<NTILE * (PROJ / 16) * 32 / 256, 256, 0, stream>>>(x, Wp, bp, xp);
  apply_bn_y<<<(MM * DIMY) / 256, 256, 0, stream>>>(y, a_y, b_y, yn);
  norms_kernel<<<(2 * MM) / 256, 256, 0, stream>>>(xp, yn, n2x, n2y);

  const int tile_blocks = NTILES2 * 32 / 256;  // 8192
  // median(pdx) -> sigma_x
  hist_hi_kernel<<<tile_blocks, 256, 0, stream>>>(xp, n2x, hist1);
  scan_hi_kernel<<<1, 256, 0, stream>>>(hist1, RANK1, RANK2, sel);
  hist_lo_kernel<<<tile_blocks, 256, 0, stream>>>(xp, n2x, sel, h2a, h2b);
  scan_lo_kernel<<<1, 256, 0, stream>>>(h2a, h2b, sel, sigx);
  // median(pdy) -> sigma_y
  zero_kernel<<<256, 256, 0, stream>>>(hist1, hist_words);
  hist_hi_kernel<<<tile_blocks, 256, 0, stream>>>(yn, n2y, hist1);
  scan_hi_kernel<<<1, 256, 0, stream>>>(hist1, RANK1, RANK2, sel);
  hist_lo_kernel<<<tile_blocks, 256, 0, stream>>>(yn, n2y, sel, h2a, h2b);
  scan_lo_kernel<<<1, 256, 0, stream>>>(h2a, h2b, sel, sigy);
  // fused HSIC pass + finalize
  hsic_pass_kernel<<<tile_blocks, 256, 0, stream>>>(
      xp, yn, n2x, n2y, sigx, sigy, Ksum, Lsum, acc3);
  finalize_kernel<<<1, 256, 0, stream>>>(Ksum, Lsum, acc3, (float*)d_out);
}